// PolicyGraphUNet_111669150313
// MI455X (gfx1250) — compile-verified
//
#include <hip/hip_runtime.h>

// ---------------------------------------------------------------------------
// Graph U-Net (dense adjacency) for MI455X / gfx1250.
// All heavy matmuls routed through V_WMMA_F32_16X16X4_F32 (fp32 matrix path).
// ---------------------------------------------------------------------------

typedef float v2f __attribute__((ext_vector_type(2)));
typedef float v8f __attribute__((ext_vector_type(8)));

#define BMt 128        // block tile M
#define BNt 64         // block tile N
#define BK  16         // staged K chunk

// ---------------------------------------------------------------------------
// Generic WMMA GEMM:  C[M x Nc] = A[M x K] @ B[K x Nc] (+bias[Nc]) (+ReLU)
// 256 threads = 8 waves arranged 4 (M) x 2 (N); each wave owns a 32x32 slice
// -> 2x2 grid of 16x16 f32 WMMA accumulators (A/B fragment reuse = 2x each).
// ---------------------------------------------------------------------------
__global__ __launch_bounds__(256)
void wmma_gemm_f32(const float* __restrict__ A, const float* __restrict__ B,
                   float* __restrict__ C, const float* __restrict__ bias,
                   int M, int K, int Nc, int relu)
{
    __shared__ float As[BMt][BK + 1];   // +1 pad: kills bank conflicts on A frag reads
    __shared__ float Bs[BK][BNt + 1];

    const int tid  = threadIdx.x;
    const int wave = tid >> 5;
    const int lane = tid & 31;
    const int wm   = (wave >> 1) * 32;  // wave M offset in tile: 0,32,64,96
    const int wn   = (wave & 1) * 32;   // wave N offset in tile: 0,32
    const int bm   = blockIdx.x * BMt;
    const int bn   = blockIdx.y * BNt;

    const int lhalf = lane & 15;        // 0..15
    const int lhi   = lane >> 4;        // 0 or 1

    v8f acc00 = {}, acc01 = {}, acc10 = {}, acc11 = {};

    // uniform across the block: does this tile need M/N edge predication?
    const bool interior = (bm + BMt <= M) && (bn + BNt <= Nc);

    for (int k0 = 0; k0 < K; k0 += BK) {
        const bool fullK = (k0 + BK <= K);

        if (interior && fullK) {
            // ---- fast path: branch-free staging ----
#pragma unroll
            for (int i = 0; i < 8; ++i) {           // A: 128x16 = 2048, 8/thread
                int e = tid * 8 + i;
                int r = e >> 4, c = e & 15;
                As[r][c] = A[(long)(bm + r) * K + (k0 + c)];
            }
#pragma unroll
            for (int i = 0; i < 4; ++i) {           // B: 16x64 = 1024, 4/thread
                int e = tid * 4 + i;
                int r = e >> 6, c = e & 63;
                Bs[r][c] = B[(long)(k0 + r) * Nc + (bn + c)];
            }
            if (k0 + 2 * BK <= K) {                 // uniform-guard prefetch of next chunk
                __builtin_prefetch(&A[(long)(bm + (tid >> 1)) * K + (k0 + BK)], 0, 0);
                __builtin_prefetch(&B[(long)(k0 + BK + (tid >> 6)) * Nc + (bn + (tid & 63))], 0, 0);
            }
        } else {
            // ---- edge path: predicated staging, zero-fill OOB ----
#pragma unroll
            for (int i = 0; i < 8; ++i) {
                int e = tid * 8 + i;
                int r = e >> 4, c = e & 15;
                int gr = bm + r, gc = k0 + c;
                float v = 0.0f;
                if (gr < M && gc < K) v = A[(long)gr * K + gc];
                As[r][c] = v;
            }
#pragma unroll
            for (int i = 0; i < 4; ++i) {
                int e = tid * 4 + i;
                int r = e >> 6, c = e & 63;
                int gr = k0 + r, gc = bn + c;
                float v = 0.0f;
                if (gr < K && gc < Nc) v = B[(long)gr * Nc + gc];
                Bs[r][c] = v;
            }
        }
        __syncthreads();

        // ---- 4 K-substeps of 4 -> 16 WMMAs per staged chunk per wave ----
#pragma unroll
        for (int s = 0; s < 4; ++s) {
            const int ks = s * 4;
            // A fragments 16x4 (ISA: lanes 0-15 K={0,1}, lanes 16-31 K={2,3})
            v2f a0, a1;
            a0[0] = As[wm + lhalf][ks + 2 * lhi + 0];
            a0[1] = As[wm + lhalf][ks + 2 * lhi + 1];
            a1[0] = As[wm + 16 + lhalf][ks + 2 * lhi + 0];
            a1[1] = As[wm + 16 + lhalf][ks + 2 * lhi + 1];
            // B fragments 4x16
            v2f b0, b1;
            b0[0] = Bs[ks + 2 * lhi + 0][wn + lhalf];
            b0[1] = Bs[ks + 2 * lhi + 1][wn + lhalf];
            b1[0] = Bs[ks + 2 * lhi + 0][wn + 16 + lhalf];
            b1[1] = Bs[ks + 2 * lhi + 1][wn + 16 + lhalf];

            acc00 = __builtin_amdgcn_wmma_f32_16x16x4_f32(false, a0, false, b0, (short)0, acc00, false, false);
            acc01 = __builtin_amdgcn_wmma_f32_16x16x4_f32(false, a0, false, b1, (short)0, acc01, false, false);
            acc10 = __builtin_amdgcn_wmma_f32_16x16x4_f32(false, a1, false, b0, (short)0, acc10, false, false);
            acc11 = __builtin_amdgcn_wmma_f32_16x16x4_f32(false, a1, false, b1, (short)0, acc11, false, false);
        }
        __syncthreads();
    }

    // ---- epilogue: C/D layout -> VGPR r holds M = r (+8 for lane>=16) ----
    const int col0 = bn + wn + lhalf;
    const int col1 = col0 + 16;
    const float bia0 = bias ? ((col0 < Nc) ? bias[col0] : 0.0f) : 0.0f;
    const float bia1 = bias ? ((col1 < Nc) ? bias[col1] : 0.0f) : 0.0f;
#pragma unroll
    for (int r = 0; r < 8; ++r) {
        int row0 = bm + wm + r + 8 * lhi;
        int row1 = row0 + 16;
        if (interior) {
            float v;
            v = acc00[r] + bia0; C[(long)row0 * Nc + col0] = relu ? fmaxf(v, 0.0f) : v;
            v = acc01[r] + bia1; C[(long)row0 * Nc + col1] = relu ? fmaxf(v, 0.0f) : v;
            v = acc10[r] + bia0; C[(long)row1 * Nc + col0] = relu ? fmaxf(v, 0.0f) : v;
            v = acc11[r] + bia1; C[(long)row1 * Nc + col1] = relu ? fmaxf(v, 0.0f) : v;
        } else {
            if (row0 < M && col0 < Nc) { float v = acc00[r] + bia0; C[(long)row0 * Nc + col0] = relu ? fmaxf(v, 0.0f) : v; }
            if (row0 < M && col1 < Nc) { float v = acc01[r] + bia1; C[(long)row0 * Nc + col1] = relu ? fmaxf(v, 0.0f) : v; }
            if (row1 < M && col0 < Nc) { float v = acc10[r] + bia0; C[(long)row1 * Nc + col0] = relu ? fmaxf(v, 0.0f) : v; }
            if (row1 < M && col1 < Nc) { float v = acc11[r] + bia1; C[(long)row1 * Nc + col1] = relu ? fmaxf(v, 0.0f) : v; }
        }
    }
}

// ---------------------------------------------------------------------------
// Elementwise / helper kernels
// ---------------------------------------------------------------------------
__global__ __launch_bounds__(256)
void zero_f32(float* __restrict__ p, long n) {
    long i = (long)blockIdx.x * 256 + threadIdx.x;
    if (i < n) p[i] = 0.0f;
}

__global__ __launch_bounds__(256)
void copy_f32(const float* __restrict__ s, float* __restrict__ d, long n) {
    long i = (long)blockIdx.x * 256 + threadIdx.x;
    if (i < n) d[i] = s[i];
}

__global__ __launch_bounds__(256)
void scatter_edges(const int* __restrict__ ei, const float* __restrict__ ew,
                   float* __restrict__ A, int E, int n) {
    int e = blockIdx.x * 256 + threadIdx.x;
    if (e < E) {
        int u = ei[e];
        int v = ei[E + e];
        A[(long)u * n + v] = ew[e];
    }
}

// dis[v] = rsqrt(2 + sum_u A[u,v])   (deg of A_hat = A + 2I, always > 0)
__global__ __launch_bounds__(256)
void col_degree_rsqrt(const float* __restrict__ A, float* __restrict__ dis, int n) {
    int v = blockIdx.x * 256 + threadIdx.x;
    if (v < n) {
        float s = 2.0f;
        for (int u = 0; u < n; ++u) s += A[(long)u * n + v];   // coalesced across v
        dis[v] = (s > 0.0f) ? rsqrtf(s) : 0.0f;
    }
}

// AnT[v,u] = dis[v]*dis[u]*(A[u,v] + 2*(u==v))  == A_norm^T
__global__ __launch_bounds__(256)
void build_norm_adj_T(const float* __restrict__ A, const float* __restrict__ dis,
                      float* __restrict__ AnT, int n) {
    long idx = (long)blockIdx.x * 256 + threadIdx.x;
    if (idx < (long)n * n) {
        int v = (int)(idx / n);
        int u = (int)(idx % n);
        float a = A[(long)u * n + v] + ((u == v) ? 2.0f : 0.0f);
        AnT[idx] = dis[v] * dis[u] * a;
    }
}

__global__ __launch_bounds__(256)
void add_identity(const float* __restrict__ A, float* __restrict__ Ap, int n) {
    long idx = (long)blockIdx.x * 256 + threadIdx.x;
    if (idx < (long)n * n) {
        int r = (int)(idx / n);
        int c = (int)(idx % n);
        Ap[idx] = A[idx] + ((r == c) ? 1.0f : 0.0f);
    }
}

// score[v] = tanh( (h[v,:] . w) / ||w|| ),  H == 128
__global__ __launch_bounds__(256)
void pool_score(const float* __restrict__ h, const float* __restrict__ w,
                float* __restrict__ score, int n) {
    int v = blockIdx.x * 256 + threadIdx.x;
    if (v < n) {
        float nw = 0.0f;
#pragma unroll 8
        for (int f = 0; f < 128; ++f) nw += w[f] * w[f];
        float s = 0.0f;
#pragma unroll 8
        for (int f = 0; f < 128; ++f) s += h[(long)v * 128 + f] * w[f];
        score[v] = tanhf(s * rsqrtf(nw));
    }
}

// Stable exact top-k (matches jax.lax.top_k: descending, ties by lower index)
__global__ __launch_bounds__(256)
void topk_perm(const float* __restrict__ score, int* __restrict__ perm,
               float* __restrict__ vals, int n, int k) {
    int i = blockIdx.x * 256 + threadIdx.x;
    if (i < n) {
        float si = score[i];
        int rank = 0;
        for (int j = 0; j < n; ++j) {
            float sj = score[j];
            rank += (sj > si) || (sj == si && j < i);
        }
        if (rank < k) {
            perm[rank] = i;
            vals[rank] = si;
        }
    }
}

// out[r,:] = h[perm[r],:] * vals[r]
__global__ __launch_bounds__(256)
void gather_scale(const float* __restrict__ h, const int* __restrict__ perm,
                  const float* __restrict__ vals, float* __restrict__ out, int k) {
    long idx = (long)blockIdx.x * 256 + threadIdx.x;
    if (idx < (long)k * 128) {
        int r = (int)(idx >> 7);
        int f = (int)(idx & 127);
        out[idx] = h[(long)perm[r] * 128 + f] * vals[r];
    }
}

// out[r,c] = (r==c) ? 0 : A2[perm[r], perm[c]]   (diag removal folded in)
__global__ __launch_bounds__(256)
void permute_adj(const float* __restrict__ A2, const int* __restrict__ perm,
                 float* __restrict__ out, int n, int k) {
    long idx = (long)blockIdx.x * 256 + threadIdx.x;
    if (idx < (long)k * k) {
        int r = (int)(idx / k);
        int c = (int)(idx % k);
        out[idx] = (r == c) ? 0.0f : A2[(long)perm[r] * n + perm[c]];
    }
}

// out[perm[r],:] += h[r,:]   (perm rows unique -> no atomics needed)
__global__ __launch_bounds__(256)
void scatter_add_rows(float* __restrict__ out, const int* __restrict__ perm,
                      const float* __restrict__ h, int k) {
    long idx = (long)blockIdx.x * 256 + threadIdx.x;
    if (idx < (long)k * 128) {
        int r = (int)(idx >> 7);
        int f = (int)(idx & 127);
        out[(long)perm[r] * 128 + f] += h[idx];
    }
}

// logits[v] = relu(h[v,:64]) . fc_w + fc_b, masked to -1e30
__global__ __launch_bounds__(256)
void final_logits(const float* __restrict__ h, const float* __restrict__ fc_w,
                  const float* __restrict__ fc_b, const unsigned char* __restrict__ mask,
                  float* __restrict__ logits, int n) {
    int v = blockIdx.x * 256 + threadIdx.x;
    if (v < n) {
        float s = fc_b[0];
#pragma unroll 8
        for (int f = 0; f < 64; ++f) s += fmaxf(h[(long)v * 64 + f], 0.0f) * fc_w[f];
        logits[v] = mask[v] ? s : -1e30f;
    }
}

__global__ __launch_bounds__(1024)
void masked_softmax(const float* __restrict__ logits, const unsigned char* __restrict__ mask,
                    float* __restrict__ out, int n) {
    __shared__ float red[1024];
    int tid = threadIdx.x;
    float m = -3.4e38f;
    for (int i = tid; i < n; i += 1024) m = fmaxf(m, logits[i]);
    red[tid] = m; __syncthreads();
    for (int s = 512; s > 0; s >>= 1) {
        if (tid < s) red[tid] = fmaxf(red[tid], red[tid + s]);
        __syncthreads();
    }
    float gm = red[0]; __syncthreads();
    float sum = 0.0f;
    for (int i = tid; i < n; i += 1024) sum += __expf(logits[i] - gm);
    red[tid] = sum; __syncthreads();
    for (int s = 512; s > 0; s >>= 1) {
        if (tid < s) red[tid] += red[tid + s];
        __syncthreads();
    }
    float gs = red[0];
    for (int i = tid; i < n; i += 1024)
        out[i] = mask[i] ? (__expf(logits[i] - gm) / gs) : 0.0f;
}

// ---------------------------------------------------------------------------
// Host-side orchestration
// ---------------------------------------------------------------------------
static inline int nblk(long n) { return (int)((n + 255) / 256); }

static void launch_gemm(const float* A, const float* B, float* C, const float* bias,
                        int M, int K, int Nc, int relu, hipStream_t s) {
    dim3 g((M + BMt - 1) / BMt, (Nc + BNt - 1) / BNt);
    wmma_gemm_f32<<<g, 256, 0, s>>>(A, B, C, bias, M, K, Nc, relu);
}

// GCN: out = AnT @ (h @ W) + b   (AnT built from Aadj + 2I, sym-norm)
static void run_gcn(const float* h, const float* Aadj, int n,
                    const float* W, const float* b, int Wout, int relu,
                    float* out, float* dis, float* AnT, float* tmpT, hipStream_t s) {
    col_degree_rsqrt<<<nblk(n), 256, 0, s>>>(Aadj, dis, n);
    build_norm_adj_T<<<nblk((long)n * n), 256, 0, s>>>(Aadj, dis, AnT, n);
    launch_gemm(h, W, tmpT, nullptr, n, 128, Wout, 0, s);
    launch_gemm(AnT, tmpT, out, b, n, n, Wout, relu, s);
}

extern "C" void kernel_launch(void* const* d_in, const int* in_sizes, int n_in,
                              void* d_out, int out_size, void* d_ws, size_t ws_size,
                              hipStream_t stream) {
    const int N = 3000, E = 48000, H = 128, OUT = 64;
    const int K1 = 1500, K2 = 750, K3 = 375;

    const float*         x         = (const float*)d_in[0];
    const int*           ei        = (const int*)d_in[1];
    const float*         ew        = (const float*)d_in[2];
    const unsigned char* mask      = (const unsigned char*)d_in[3];
    const float*         down_w    = (const float*)d_in[4];   // [4,128,128]
    const float*         down_b    = (const float*)d_in[5];   // [4,128]
    const float*         pool_w    = (const float*)d_in[6];   // [3,128]
    const float*         up_w      = (const float*)d_in[7];   // [2,128,128]
    const float*         up_b      = (const float*)d_in[8];   // [2,128]
    const float*         up_w_last = (const float*)d_in[9];   // [128,64]
    const float*         up_b_last = (const float*)d_in[10];  // [64]
    const float*         fc_w      = (const float*)d_in[11];  // [64,1]
    const float*         fc_b      = (const float*)d_in[12];  // [1]

    // ---- workspace layout (floats) ----
    float* base = (float*)d_ws;
    size_t off = 0;
    auto alloc = [&](size_t n) { float* p = base + off; off += n; return p; };
    float* A0   = alloc((size_t)N * N);       // adjacency (kept for last up-GCN)
    float* SCR1 = alloc((size_t)N * N);       // AnT / (A+I) scratch
    float* SCR2 = alloc((size_t)N * N);       // augmented A^2 scratch
    float* A1   = alloc((size_t)K1 * K1);
    float* A2s  = alloc((size_t)K2 * K2);
    float* A3   = alloc((size_t)K3 * K3);
    float* xs0  = alloc((size_t)N * H);
    float* xs1  = alloc((size_t)K1 * H);
    float* xs2  = alloc((size_t)K2 * H);
    float* tmpT = alloc((size_t)N * H);       // GEMM intermediate h@W
    float* hcur = alloc((size_t)K1 * H);      // permuted h (max 1500x128); reused as g0
    float* hd3  = alloc((size_t)K3 * H);      // bottom of U
    float* g1   = alloc((size_t)K1 * H);      // up-GCN i=1 output
    float* hu0  = alloc((size_t)K2 * H);
    float* hu1  = alloc((size_t)K1 * H);
    float* hu2  = alloc((size_t)N * H);
    float* hfin = alloc((size_t)N * OUT);
    float* dis  = alloc(N);
    float* score= alloc(N);
    float* vals0= alloc(K1);
    float* vals1= alloc(K2);
    float* vals2= alloc(K3);
    float* logit= alloc(N);
    int* p0 = (int*)(base + off); off += K1;
    int* p1 = (int*)(base + off); off += K2;
    int* p2 = (int*)(base + off); off += K3;
    (void)ws_size; (void)in_sizes; (void)n_in; (void)out_size;

    // ---- build dense adjacency ----
    zero_f32<<<nblk((long)N * N), 256, 0, stream>>>(A0, (long)N * N);
    scatter_edges<<<nblk(E), 256, 0, stream>>>(ei, ew, A0, E, N);

    // ---- first GCN: h0 = relu(gcn(x, A0, down_w[0])) ----
    run_gcn(x, A0, N, down_w, down_b, H, 1, xs0, dis, SCR1, tmpT, stream);

    // ---- down level 0: 3000 -> 1500 ----
    add_identity<<<nblk((long)N * N), 256, 0, stream>>>(A0, SCR1, N);
    launch_gemm(SCR1, SCR1, SCR2, nullptr, N, N, N, 0, stream);
    pool_score<<<nblk(N), 256, 0, stream>>>(xs0, pool_w, score, N);
    topk_perm<<<nblk(N), 256, 0, stream>>>(score, p0, vals0, N, K1);
    gather_scale<<<nblk((long)K1 * H), 256, 0, stream>>>(xs0, p0, vals0, hcur, K1);
    permute_adj<<<nblk((long)K1 * K1), 256, 0, stream>>>(SCR2, p0, A1, N, K1);
    run_gcn(hcur, A1, K1, down_w + 1 * H * H, down_b + 1 * H, H, 1, xs1, dis, SCR1, tmpT, stream);

    // ---- down level 1: 1500 -> 750 ----
    add_identity<<<nblk((long)K1 * K1), 256, 0, stream>>>(A1, SCR1, K1);
    launch_gemm(SCR1, SCR1, SCR2, nullptr, K1, K1, K1, 0, stream);
    pool_score<<<nblk(K1), 256, 0, stream>>>(xs1, pool_w + H, score, K1);
    topk_perm<<<nblk(K1), 256, 0, stream>>>(score, p1, vals1, K1, K2);
    gather_scale<<<nblk((long)K2 * H), 256, 0, stream>>>(xs1, p1, vals1, hcur, K2);
    permute_adj<<<nblk((long)K2 * K2), 256, 0, stream>>>(SCR2, p1, A2s, K1, K2);
    run_gcn(hcur, A2s, K2, down_w + 2 * H * H, down_b + 2 * H, H, 1, xs2, dis, SCR1, tmpT, stream);

    // ---- down level 2: 750 -> 375 ----
    add_identity<<<nblk((long)K2 * K2), 256, 0, stream>>>(A2s, SCR1, K2);
    launch_gemm(SCR1, SCR1, SCR2, nullptr, K2, K2, K2, 0, stream);
    pool_score<<<nblk(K2), 256, 0, stream>>>(xs2, pool_w + 2 * H, score, K2);
    topk_perm<<<nblk(K2), 256, 0, stream>>>(score, p2, vals2, K2, K3);
    gather_scale<<<nblk((long)K3 * H), 256, 0, stream>>>(xs2, p2, vals2, hcur, K3);
    permute_adj<<<nblk((long)K3 * K3), 256, 0, stream>>>(SCR2, p2, A3, K2, K3);
    run_gcn(hcur, A3, K3, down_w + 3 * H * H, down_b + 3 * H, H, 1, hd3, dis, SCR1, tmpT, stream);

    // ---- up level 0: 375 -> 750 ----
    copy_f32<<<nblk((long)K2 * H), 256, 0, stream>>>(xs2, hu0, (long)K2 * H);
    scatter_add_rows<<<nblk((long)K3 * H), 256, 0, stream>>>(hu0, p2, hd3, K3);
    run_gcn(hu0, A2s, K2, up_w, up_b, H, 1, hcur /* g0 */, dis, SCR1, tmpT, stream);

    // ---- up level 1: 750 -> 1500 ----
    copy_f32<<<nblk((long)K1 * H), 256, 0, stream>>>(xs1, hu1, (long)K1 * H);
    scatter_add_rows<<<nblk((long)K2 * H), 256, 0, stream>>>(hu1, p1, hcur, K2);
    run_gcn(hu1, A1, K1, up_w + H * H, up_b + H, H, 1, g1, dis, SCR1, tmpT, stream);

    // ---- up level 2 (last, no relu, 128 -> 64): 1500 -> 3000 ----
    copy_f32<<<nblk((long)N * H), 256, 0, stream>>>(xs0, hu2, (long)N * H);
    scatter_add_rows<<<nblk((long)K1 * H), 256, 0, stream>>>(hu2, p0, g1, K1);
    run_gcn(hu2, A0, N, up_w_last, up_b_last, OUT, 0, hfin, dis, SCR1, tmpT, stream);

    // ---- readout: relu -> fc -> masked softmax ----
    final_logits<<<nblk(N), 256, 0, stream>>>(hfin, fc_w, fc_b, mask, logit, N);
    masked_softmax<<<1, 1024, 0, stream>>>(logit, mask, (float*)d_out, N);
}